// NystromAttention_86122684219922
// MI455X (gfx1250) — compile-verified
//
#include <hip/hip_runtime.h>

#define B_ 8
#define H_ 8
#define N_ 4096
#define D_ 64
#define M_ 64
#define L_ 64
#define KER_ 33
#define PADK_ 16
#define LDP 68  // padded LDS row stride: 16B-aligned rows, conflict-free column reads

typedef __attribute__((ext_vector_type(2))) float v2f;
typedef __attribute__((ext_vector_type(8))) float v8f;
typedef int v4i_ __attribute__((vector_size(16)));

#define AS1 __attribute__((address_space(1)))
#define AS3 __attribute__((address_space(3)))

#if __has_builtin(__builtin_amdgcn_global_load_async_to_lds_b128)
#define HAVE_ASYNC 1
#else
#define HAVE_ASYNC 0
#endif

__device__ __forceinline__ AS1 v4i_* glb128(const void* p) {
  return (AS1 v4i_*)(unsigned long long)p;
}
__device__ __forceinline__ AS3 v4i_* lds128(void* p) {
  return (AS3 v4i_*)(unsigned int)(unsigned long long)p;
}

__device__ __forceinline__ void tile_wait() {
#if HAVE_ASYNC
#if __has_builtin(__builtin_amdgcn_s_wait_asynccnt)
  __builtin_amdgcn_s_wait_asynccnt(0);
#else
  asm volatile("s_wait_asynccnt 0x0" ::: "memory");
#endif
#endif
}

// Copy a 64x64 f32 tile (row stride g_stride floats) from global into LDS
// (row stride LDP floats). Async DMA path on CDNA5, VGPR fallback otherwise.
__device__ __forceinline__ void load_tile64(const float* __restrict__ g,
                                            int g_stride, float* lds) {
#if HAVE_ASYNC
  for (int idx = threadIdx.x; idx < 1024; idx += blockDim.x) {
    const int r  = idx >> 4;
    const int c4 = (idx & 15) << 2;
    __builtin_amdgcn_global_load_async_to_lds_b128(
        glb128(g + (size_t)r * g_stride + c4), lds128(lds + r * LDP + c4), 0, 0);
  }
#else
  for (int idx = threadIdx.x; idx < 4096; idx += blockDim.x) {
    const int r = idx >> 6, c = idx & 63;
    lds[r * LDP + c] = g[(size_t)r * g_stride + c];
  }
#endif
}

// ---------------- WMMA f32 16x16x4 helpers ----------------

__device__ __forceinline__ v8f wmma_step(v2f a, v2f b, v8f c) {
  return __builtin_amdgcn_wmma_f32_16x16x4_f32(false, a, false, b, (short)0, c,
                                               false, false);
}

// C tile (ti,tj) += A[64xK] @ B[Kx64]
__device__ __forceinline__ v8f tile_AB(const float* __restrict__ A, int lda,
                                       const float* __restrict__ Bm, int ldb,
                                       int ti, int tj, int K, v8f c) {
  const int lane = threadIdx.x & 31;
  const int half = lane >> 4;
  const int lid  = lane & 15;
  const int ar = (ti * 16 + lid) * lda;
  const int bc = tj * 16 + lid;
#pragma unroll 4
  for (int kk = 0; kk < K; kk += 4) {
    const int k0 = kk + 2 * half;
    v2f a, b;
    a.x = A[ar + k0];
    a.y = A[ar + k0 + 1];
    b.x = Bm[k0 * ldb + bc];
    b.y = Bm[(k0 + 1) * ldb + bc];
    c = wmma_step(a, b, c);
  }
  return c;
}

// C tile (ti,tj) += A[64xK] @ Bt[64xK]^T
__device__ __forceinline__ v8f tile_ABt(const float* __restrict__ A, int lda,
                                        const float* __restrict__ Bt, int ldb,
                                        int ti, int tj, int K, v8f c) {
  const int lane = threadIdx.x & 31;
  const int half = lane >> 4;
  const int lid  = lane & 15;
  const int ar = (ti * 16 + lid) * lda;
  const int br = (tj * 16 + lid) * ldb;
#pragma unroll 4
  for (int kk = 0; kk < K; kk += 4) {
    const int k0 = kk + 2 * half;
    v2f a, b;
    a.x = A[ar + k0];
    a.y = A[ar + k0 + 1];
    b.x = Bt[br + k0];
    b.y = Bt[br + k0 + 1];
    c = wmma_step(a, b, c);
  }
  return c;
}

__device__ __forceinline__ void tile_store(float* __restrict__ C, int ldc,
                                           int ti, int tj, v8f c, float scale) {
  const int lane = threadIdx.x & 31;
  const int half = lane >> 4;
  const int lid  = lane & 15;
#pragma unroll
  for (int r = 0; r < 8; ++r)
    C[(ti * 16 + r + 8 * half) * ldc + tj * 16 + lid] = c[r] * scale;
}

// ---------------- K1: landmark means ----------------
__global__ void k_landmarks(const float* __restrict__ q, const float* __restrict__ k,
                            float* __restrict__ ql, float* __restrict__ kl) {
  const int bh = blockIdx.x >> 6;
  const int mr = blockIdx.x & 63;
  const int d  = threadIdx.x;  // 64 threads
  const size_t base = ((size_t)bh * N_ + (size_t)mr * L_) * D_ + d;
  float sq = 0.f, sk = 0.f;
  for (int r = 0; r < L_; ++r) {
    sq += q[base + (size_t)r * D_];
    sk += k[base + (size_t)r * D_];
  }
  const size_t o = ((size_t)bh * M_ + mr) * D_ + d;
  ql[o] = sq * (1.0f / L_);
  kl[o] = sk * (1.0f / L_);
}

// ---------------- K2: sim2 = q_l @ k_l^T ----------------
__global__ void __launch_bounds__(512) k_sim2(const float* __restrict__ ql,
                                              const float* __restrict__ kl,
                                              float* __restrict__ sim2) {
  __shared__ float qs[M_ * LDP], ks[M_ * LDP];
  const int bh  = blockIdx.x;
  const int tid = threadIdx.x;
  const size_t base = (size_t)bh * M_ * D_;
  load_tile64(ql + base, D_, qs);
  load_tile64(kl + base, D_, ks);
  tile_wait();
  __syncthreads();
  const int w = tid >> 5, ti = w >> 2, tj = w & 3;
  v8f c = {};
  c = tile_ABt(qs, LDP, ks, LDP, ti, tj, D_, c);
  tile_store(sim2 + base, M_, ti, tj, c, 1.f);
}

// ---------------- K3: sim3 (q_l @ k^T) fused with Wsim3 head mix ----------------
__global__ void __launch_bounds__(512) k_sim3_mix(const float* __restrict__ ql,
                                                  const float* __restrict__ kg,
                                                  const float* __restrict__ Wsim3,
                                                  float* __restrict__ sim3m) {
  __shared__ float as[M_ * LDP], bs[M_ * LDP];
  const int b   = blockIdx.y;
  const int j0  = blockIdx.x * 64;
  const int tid = threadIdx.x;
  const int w = tid >> 5, ti = w >> 2, tj = w & 3;
  v8f acc[H_] = {};
  for (int h = 0; h < H_; ++h) {
    const size_t qb = (size_t)(b * H_ + h) * M_ * D_;
    const size_t kb = ((size_t)(b * H_ + h) * N_ + j0) * D_;
    load_tile64(ql + qb, D_, as);
    load_tile64(kg + kb, D_, bs);
    tile_wait();
    __syncthreads();
    v8f c = {};
    c = tile_ABt(as, LDP, bs, LDP, ti, tj, D_, c);
#pragma unroll
    for (int kk = 0; kk < H_; ++kk) acc[kk] += c * Wsim3[h * H_ + kk];
    __syncthreads();
  }
  const int lane = tid & 31, half = lane >> 4, lid = lane & 15;
  for (int kk = 0; kk < H_; ++kk) {
    float* ob = sim3m + (size_t)(b * H_ + kk) * M_ * (size_t)N_ + j0;
#pragma unroll
    for (int r = 0; r < 8; ++r)
      ob[(size_t)(ti * 16 + r + 8 * half) * N_ + tj * 16 + lid] = acc[kk][r];
  }
}

// ---------------- K4: per-row softmax stats (max, sumexp) over N ----------------
__global__ void k_rowstats(const float* __restrict__ sm, float* __restrict__ stats) {
  __shared__ float red[256];
  const int row = blockIdx.x;  // B*H*M rows
  const float* p = sm + (size_t)row * N_;
  const int tid = threadIdx.x;
  float mx = -3.4e38f;
  for (int j = tid; j < N_; j += 256) mx = fmaxf(mx, p[j]);
  red[tid] = mx; __syncthreads();
  for (int s = 128; s > 0; s >>= 1) {
    if (tid < s) red[tid] = fmaxf(red[tid], red[tid + s]);
    __syncthreads();
  }
  mx = red[0]; __syncthreads();
  float sum = 0.f;
  for (int j = tid; j < N_; j += 256) sum += __expf(p[j] - mx);
  red[tid] = sum; __syncthreads();
  for (int s = 128; s > 0; s >>= 1) {
    if (tid < s) red[tid] += red[tid + s];
    __syncthreads();
  }
  if (tid == 0) { stats[row * 2] = mx; stats[row * 2 + 1] = red[0]; }
}

// ---------------- K5: softmax-normalize + Wattn3 head mix (in place) ----------------
__global__ void k_attn3_mix(float* __restrict__ sm, const float* __restrict__ stats,
                            const float* __restrict__ Wattn3) {
  __shared__ float wl[64];
  const int tid = threadIdx.x;
  if (tid < 64) wl[tid] = Wattn3[tid];
  __syncthreads();
  const size_t idx = (size_t)blockIdx.x * 256 + tid;  // < B*M*N
  const int j = (int)(idx & (N_ - 1));
  const int i = (int)((idx >> 12) & (M_ - 1));
  const int b = (int)(idx >> 18);
  float val[H_], outv[H_];
#pragma unroll
  for (int h = 0; h < H_; ++h) {
    const int row = (b * H_ + h) * M_ + i;
    float s = sm[(size_t)row * N_ + j];
    val[h] = __expf(s - stats[row * 2]) / stats[row * 2 + 1];
  }
#pragma unroll
  for (int kk = 0; kk < H_; ++kk) {
    float o = 0.f;
#pragma unroll
    for (int h = 0; h < H_; ++h) o += val[h] * wl[h * H_ + kk];
    outv[kk] = o;
  }
#pragma unroll
  for (int kk = 0; kk < H_; ++kk)
    sm[((size_t)(b * H_ + kk) * M_ + i) * N_ + j] = outv[kk];
}

// ---------------- K6: t3 = attn3m @ v, double-buffered async pipeline ----------------
__global__ void __launch_bounds__(512) k_t3(const float* __restrict__ am,
                                            const float* __restrict__ v,
                                            float* __restrict__ t3) {
  extern __shared__ float smem[];
  const int bh  = blockIdx.x;
  const int tid = threadIdx.x;
  const int w = tid >> 5, ti = w >> 2, tj = w & 3;
  const float* ab = am + (size_t)bh * M_ * N_;
  const float* vb = v + (size_t)bh * N_ * D_;
  v8f c = {};
  load_tile64(ab, N_, smem);
  load_tile64(vb, D_, smem + 2 * 64 * LDP);
  tile_wait();
  __syncthreads();
  for (int kc = 0; kc < N_; kc += 64) {
    const int cur = (kc >> 6) & 1, nxt = cur ^ 1;
    float* aCur = smem + cur * 64 * LDP;
    float* vCur = smem + (2 + cur) * 64 * LDP;
    if (kc + 64 < N_) {
      load_tile64(ab + kc + 64, N_, smem + nxt * 64 * LDP);
      load_tile64(vb + (size_t)(kc + 64) * D_, D_, smem + (2 + nxt) * 64 * LDP);
    }
    c = tile_AB(aCur, LDP, vCur, LDP, ti, tj, 64, c);
    tile_wait();
    __syncthreads();
  }
  tile_store(t3 + (size_t)bh * M_ * D_, D_, ti, tj, c, 1.f);
}

// ---------------- K7a: attn2 = softmax(mix(sim2, Wsim2)) ----------------
__global__ void k_attn2(const float* __restrict__ sim2, const float* __restrict__ Wsim2,
                        float* __restrict__ attn2) {
  __shared__ float s[64 * LDP];
  const int b  = blockIdx.x >> 3;
  const int kp = blockIdx.x & 7;
  const int tid = threadIdx.x;  // 256
  for (int idx = tid; idx < 64 * 64; idx += 256) {
    int i = idx >> 6, j = idx & 63;
    float o = 0.f;
#pragma unroll
    for (int h = 0; h < H_; ++h)
      o += sim2[((size_t)(b * H_ + h) * M_ + i) * M_ + j] * Wsim2[h * H_ + kp];
    s[i * LDP + j] = o;
  }
  __syncthreads();
  if (tid < 64) {
    float* row = s + tid * LDP;
    float mx = row[0];
    for (int j = 1; j < 64; ++j) mx = fmaxf(mx, row[j]);
    float sum = 0.f;
    for (int j = 0; j < 64; ++j) { float e = __expf(row[j] - mx); row[j] = e; sum += e; }
    float inv = 1.f / sum;
    for (int j = 0; j < 64; ++j) row[j] *= inv;
  }
  __syncthreads();
  float* ob = attn2 + (size_t)(b * H_ + kp) * M_ * M_;
  for (int idx = tid; idx < 64 * 64; idx += 256)
    ob[idx] = s[(idx >> 6) * LDP + (idx & 63)];
}

// ---------------- K7b: Moore-Penrose iterative pinv (all WMMA in LDS) ----------------
__global__ void __launch_bounds__(512) k_pinv(const float* __restrict__ attn2,
                                              float* __restrict__ pv) {
  extern __shared__ float sm[];
  float* xs  = sm;
  float* zs  = xs + 64 * LDP;
  float* xzs = zs + 64 * LDP;
  float* ts  = xzs + 64 * LDP;
  float* red = ts + 64 * LDP;  // 130 floats
  const int bh  = blockIdx.x;
  const int tid = threadIdx.x;
  const int w = tid >> 5, ti = w >> 2, tj = w & 3;
  load_tile64(attn2 + (size_t)bh * M_ * M_, M_, xs);
  tile_wait();
  __syncthreads();
  if (tid < 64) {
    float cs = 0.f;
    for (int j = 0; j < 64; ++j) cs += fabsf(xs[tid * LDP + j]);
    red[tid] = cs;
  } else if (tid < 128) {
    int j = tid - 64;
    float rs = 0.f;
    for (int i = 0; i < 64; ++i) rs += fabsf(xs[i * LDP + j]);
    red[64 + j] = rs;
  }
  __syncthreads();
  if (tid == 0) {
    float mc = red[0], mr = red[64];
    for (int i = 1; i < 64; ++i) { mc = fmaxf(mc, red[i]); mr = fmaxf(mr, red[64 + i]); }
    red[128] = 1.f / (mc * mr);
  }
  __syncthreads();
  const float sc = red[128];
  for (int idx = tid; idx < 4096; idx += 512) {
    int i = idx >> 6, j = idx & 63;
    zs[i * LDP + j] = xs[j * LDP + i] * sc;
  }
  __syncthreads();
  for (int it = 0; it < 6; ++it) {
    {  // xz = x @ z
      v8f c = {};
      c = tile_AB(xs, LDP, zs, LDP, ti, tj, 64, c);
      tile_store(xzs, LDP, ti, tj, c, 1.f);
      __syncthreads();
    }
    for (int idx = tid; idx < 4096; idx += 512) {  // t = 7I - xz
      int i = idx >> 6, j = idx & 63;
      ts[i * LDP + j] = (i == j ? 7.f : 0.f) - xzs[i * LDP + j];
    }
    __syncthreads();
    {  // t = xz @ t
      v8f c = {};
      c = tile_AB(xzs, LDP, ts, LDP, ti, tj, 64, c);
      __syncthreads();
      tile_store(ts, LDP, ti, tj, c, 1.f);
      __syncthreads();
    }
    for (int idx = tid; idx < 4096; idx += 512) {  // t = 15I - t
      int i = idx >> 6, j = idx & 63;
      ts[i * LDP + j] = (i == j ? 15.f : 0.f) - ts[i * LDP + j];
    }
    __syncthreads();
    {  // t = xz @ t
      v8f c = {};
      c = tile_AB(xzs, LDP, ts, LDP, ti, tj, 64, c);
      __syncthreads();
      tile_store(ts, LDP, ti, tj, c, 1.f);
      __syncthreads();
    }
    for (int idx = tid; idx < 4096; idx += 512) {  // t = 13I - t
      int i = idx >> 6, j = idx & 63;
      ts[i * LDP + j] = (i == j ? 13.f : 0.f) - ts[i * LDP + j];
    }
    __syncthreads();
    {  // z = 0.25 * z @ t
      v8f c = {};
      c = tile_AB(zs, LDP, ts, LDP, ti, tj, 64, c);
      __syncthreads();
      tile_store(zs, LDP, ti, tj, c, 0.25f);
      __syncthreads();
    }
  }
  float* ob = pv + (size_t)bh * M_ * M_;
  for (int idx = tid; idx < 4096; idx += 512)
    ob[idx] = zs[(idx >> 6) * LDP + (idx & 63)];
}

// ---------------- K7c: mix attn2_inv with Wattn2 ----------------
__global__ void k_mixinv(const float* __restrict__ pv, const float* __restrict__ Wattn2,
                         float* __restrict__ pim) {
  __shared__ float wl[64];
  const int tid = threadIdx.x;
  if (tid < 64) wl[tid] = Wattn2[tid];
  __syncthreads();
  const int idx = blockIdx.x * 256 + tid;  // < B*M*M
  const int ij = idx & 4095;
  const int b  = idx >> 12;
  float s[H_], o[H_];
#pragma unroll
  for (int h = 0; h < H_; ++h) s[h] = pv[(size_t)(b * H_ + h) * 4096 + ij];
#pragma unroll
  for (int kk = 0; kk < H_; ++kk) {
    float a = 0.f;
#pragma unroll
    for (int h = 0; h < H_; ++h) a += s[h] * wl[h * H_ + kk];
    o[kk] = a;
  }
#pragma unroll
  for (int kk = 0; kk < H_; ++kk) pim[(size_t)(b * H_ + kk) * 4096 + ij] = o[kk];
}

// ---------------- K7d: t2 = attn2_inv_mixed @ t3 ----------------
__global__ void __launch_bounds__(512) k_t2(const float* __restrict__ pim,
                                            const float* __restrict__ t3,
                                            float* __restrict__ t2) {
  __shared__ float as[64 * LDP], bs[64 * LDP];
  const int bh  = blockIdx.x;
  const int tid = threadIdx.x;
  load_tile64(pim + (size_t)bh * 4096, M_, as);
  load_tile64(t3 + (size_t)bh * 4096, D_, bs);
  tile_wait();
  __syncthreads();
  const int w = tid >> 5, ti = w >> 2, tj = w & 3;
  v8f c = {};
  c = tile_AB(as, LDP, bs, LDP, ti, tj, 64, c);
  tile_store(t2 + (size_t)bh * 4096, M_, ti, tj, c, 1.f);
}

// ---------------- K8: conv residual -> d_out ----------------
__global__ void k_conv(const float* __restrict__ v, const float* __restrict__ cw,
                       float* __restrict__ out) {
  __shared__ float wl[H_ * H_ * KER_];
  const int tid = threadIdx.x;
  for (int i = tid; i < H_ * H_ * KER_; i += 256) wl[i] = cw[i];
  __syncthreads();
  const size_t idx = (size_t)blockIdx.x * 256 + tid;  // B*N*D
  const int d = (int)(idx & 63);
  const int n = (int)((idx >> 6) & (N_ - 1));
  const int b = (int)(idx >> 18);
  float acc[H_] = {};
  for (int h = 0; h < H_; ++h) {
    const float* vb = v + (size_t)(b * H_ + h) * N_ * D_ + d;
    for (int t = 0; t < KER_; ++t) {
      const int nn = n + t - PADK_;
      if (nn < 0 || nn >= N_) continue;
      const float vv = vb[(size_t)nn * D_];
#pragma unroll
      for (int kk = 0; kk < H_; ++kk) acc[kk] += vv * wl[(kk * H_ + h) * KER_ + t];
    }
  }
#pragma unroll
  for (int kk = 0; kk < H_; ++kk)
    out[((size_t)(b * H_ + kk) * N_ + n) * D_ + d] = acc[kk];
}

// ---------------- K9: fused sim1 -> mix -> softmax -> mix -> @t2 (+residual) ----------------
__global__ void __launch_bounds__(512) k_final(const float* __restrict__ q,
                                               const float* __restrict__ kl,
                                               const float* __restrict__ t2,
                                               const float* __restrict__ Wsim1,
                                               const float* __restrict__ Wattn1,
                                               float* __restrict__ out) {
  extern __shared__ float sm[];
  float* S  = sm;                    // 8 * 64*LDP
  float* as = S + 8 * 64 * LDP;      // 64*LDP
  float* bs = as + 64 * LDP;         // 64*LDP
  float* w1 = bs + 64 * LDP;         // 64
  float* w2 = w1 + 64;               // 64
  const int b  = blockIdx.y;
  const int i0 = blockIdx.x * 64;
  const int tid = threadIdx.x;
  const int w = tid >> 5, ti = w >> 2, tj = w & 3;
  if (tid < 64) w1[tid] = Wsim1[tid];
  else if (tid < 128) w2[tid - 64] = Wattn1[tid - 64];
  // S[h] = q_rows @ k_l^T
  for (int h = 0; h < H_; ++h) {
    const size_t qb = ((size_t)(b * H_ + h) * N_ + i0) * D_;
    const size_t kb = (size_t)(b * H_ + h) * M_ * D_;
    load_tile64(q + qb, D_, as);
    load_tile64(kl + kb, D_, bs);
    tile_wait();
    __syncthreads();
    v8f c = {};
    c = tile_ABt(as, LDP, bs, LDP, ti, tj, D_, c);
    tile_store(S + h * 64 * LDP, LDP, ti, tj, c, 1.f);
    __syncthreads();
  }
  // mix #1 (Wsim1), in place across h
  for (int idx = tid; idx < 4096; idx += 512) {
    int i = idx >> 6, j = idx & 63;
    float s[H_], o[H_];
#pragma unroll
    for (int h = 0; h < H_; ++h) s[h] = S[h * 64 * LDP + i * LDP + j];
#pragma unroll
    for (int kk = 0; kk < H_; ++kk) {
      float a = 0.f;
#pragma unroll
      for (int h = 0; h < H_; ++h) a += s[h] * w1[h * H_ + kk];
      o[kk] = a;
    }
#pragma unroll
    for (int kk = 0; kk < H_; ++kk) S[kk * 64 * LDP + i * LDP + j] = o[kk];
  }
  __syncthreads();
  // softmax over landmark axis (64 wide): 512 rows, one per thread
  {
    const int kk = tid >> 6, i = tid & 63;
    float* row = S + kk * 64 * LDP + i * LDP;
    float mx = row[0];
    for (int j = 1; j < 64; ++j) mx = fmaxf(mx, row[j]);
    float sum = 0.f;
    for (int j = 0; j < 64; ++j) { float e = __expf(row[j] - mx); row[j] = e; sum += e; }
    float inv = 1.f / sum;
    for (int j = 0; j < 64; ++j) row[j] *= inv;
  }
  __syncthreads();
  // mix #2 (Wattn1), in place
  for (int idx = tid; idx < 4096; idx += 512) {
    int i = idx >> 6, j = idx & 63;
    float s[H_], o[H_];
#pragma unroll
    for (int h = 0; h < H_; ++h) s[h] = S[h * 64 * LDP + i * LDP + j];
#pragma unroll
    for (int kk = 0; kk < H_; ++kk) {
      float a = 0.f;
#pragma unroll
      for (int h = 0; h < H_; ++h) a += s[h] * w2[h * H_ + kk];
      o[kk] = a;
    }
#pragma unroll
    for (int kk = 0; kk < H_; ++kk) S[kk * 64 * LDP + i * LDP + j] = o[kk];
  }
  __syncthreads();
  // out[b,kp,i0+..,:] += attn1m[kp] @ t2[b,kp]
  for (int kp = 0; kp < H_; ++kp) {
    load_tile64(t2 + (size_t)(b * H_ + kp) * 4096, D_, as);
    tile_wait();
    __syncthreads();
    v8f c = {};
    c = tile_AB(S + kp * 64 * LDP, LDP, as, LDP, ti, tj, 64, c);
    const int lane = tid & 31, half = lane >> 4, lid = lane & 15;
    float* ob = out + ((size_t)(b * H_ + kp) * N_ + i0) * D_;
#pragma unroll
    for (int r = 0; r < 8; ++r) {
      const size_t a = (size_t)(ti * 16 + r + 8 * half) * D_ + tj * 16 + lid;
      ob[a] = ob[a] + c[r];
    }
    __syncthreads();
  }
}

// ---------------- launch ----------------
extern "C" void kernel_launch(void* const* d_in, const int* in_sizes, int n_in,
                              void* d_out, int out_size, void* d_ws, size_t ws_size,
                              hipStream_t stream) {
  (void)in_sizes; (void)n_in; (void)out_size; (void)ws_size;
  const float* q      = (const float*)d_in[0];
  const float* k      = (const float*)d_in[1];
  const float* v      = (const float*)d_in[2];
  const float* Wsim1  = (const float*)d_in[3];
  const float* Wsim2  = (const float*)d_in[4];
  const float* Wsim3  = (const float*)d_in[5];
  const float* Wattn1 = (const float*)d_in[6];
  const float* Wattn2 = (const float*)d_in[7];
  const float* Wattn3 = (const float*)d_in[8];
  const float* convw  = (const float*)d_in[9];
  float* out = (float*)d_out;

  float* ws = (float*)d_ws;
  const size_t SM = (size_t)B_ * H_ * M_ * M_;  // 262144
  float* ql    = ws; ws += SM;
  float* kl    = ws; ws += SM;
  float* sim2  = ws; ws += SM;
  float* attn2 = ws; ws += SM;
  float* pv    = ws; ws += SM;
  float* pim   = ws; ws += SM;
  float* t3    = ws; ws += SM;
  float* t2    = ws; ws += SM;
  float* stats = ws; ws += (size_t)B_ * H_ * M_ * 2;
  float* sim3m = ws;  // B*H*M*N floats (64 MiB)

  k_landmarks<<<B_ * H_ * M_, 64, 0, stream>>>(q, k, ql, kl);
  k_sim2<<<B_ * H_, 512, 0, stream>>>(ql, kl, sim2);
  k_sim3_mix<<<dim3(N_ / 64, B_), 512, 0, stream>>>(ql, k, Wsim3, sim3m);
  k_rowstats<<<B_ * H_ * M_, 256, 0, stream>>>(sim3m, stats);
  k_attn3_mix<<<(B_ * M_ * N_) / 256, 256, 0, stream>>>(sim3m, stats, Wattn3);
  const size_t t3_lds = (size_t)(4 * 64 * LDP) * sizeof(float);
  k_t3<<<B_ * H_, 512, t3_lds, stream>>>(sim3m, v, t3);
  k_attn2<<<B_ * H_, 256, 0, stream>>>(sim2, Wsim2, attn2);
  const size_t pinv_lds = (size_t)(4 * 64 * LDP + 130) * sizeof(float);
  k_pinv<<<B_ * H_, 512, pinv_lds, stream>>>(attn2, pv);
  k_mixinv<<<(B_ * M_ * M_) / 256, 256, 0, stream>>>(pv, Wattn2, pim);
  k_t2<<<B_ * H_, 512, 0, stream>>>(pim, t3, t2);
  k_conv<<<(B_ * N_ * D_) / 256, 256, 0, stream>>>(v, convw, out);
  const size_t fin_lds = (size_t)(10 * 64 * LDP + 128) * sizeof(float);
  k_final<<<dim3(N_ / 64, B_), 512, fin_lds, stream>>>(q, kl, t2, Wsim1, Wattn1, out);
}